// DeductronCTC_15436112462127
// MI455X (gfx1250) — compile-verified
//
#include <hip/hip_runtime.h>
#include <hip/hip_bf16.h>

typedef __attribute__((ext_vector_type(2))) float v2f;
typedef __attribute__((ext_vector_type(8))) float v8f;
typedef __attribute__((ext_vector_type(4))) unsigned int uint32x4;
typedef __attribute__((ext_vector_type(8))) int int32x8;
typedef __attribute__((ext_vector_type(4))) int int32x4;

#define NMEM   64
#define KIN    128
#define CHUNK  512
#define SUB    128   // CHUNK / 4

__device__ __forceinline__ float sigm(float x) {
  return 1.0f / (1.0f + __expf(-x));
}

// -------------------------------------------------------------------------
// Kernel 1: left = sigmoid(in @ Wl + bl), right = sigmoid(in @ Wr + br)
//           a = left*right, b = 1-left   (fused, WMMA f32 16x16x4)
// Weights quad-packed in LDS: tW[proj][k/4][n][k%4] so a B-fragment is a
// single conflict-free ds_load_b64 (lanes 0-15 read k%4 in {0,1}, lanes
// 16-31 read {2,3} -> 64 consecutive DWORDs = all 64 banks once).
// -------------------------------------------------------------------------
extern "C" __global__ __launch_bounds__(256) void k_proj(
    const float* __restrict__ in, const float* __restrict__ Wl,
    const float* __restrict__ bl, const float* __restrict__ Wr,
    const float* __restrict__ br, float* __restrict__ Aarr,
    float* __restrict__ Barr, int N)
{
  __shared__ float tW[2 * KIN * NMEM];   // 64 KB, quad-packed
  const int tid = threadIdx.x;
  const long frame0 = (long)blockIdx.x * 128;

  for (int idx = tid; idx < KIN * NMEM; idx += 256) {
    const int k = idx >> 6, n = idx & 63;
    const int dst = ((k >> 2) << 8) + (n << 2) + (k & 3);
    tW[dst]              = Wl[idx];
    tW[KIN * NMEM + dst] = Wr[idx];
  }
  __syncthreads();

  const int lane = tid & 31;
  const int wave = tid >> 5;
  const int mrow = lane & 15;
  const int kOff = (lane >> 4) << 1;          // lanes 16-31 hold K+2
  const long rowA = frame0 + wave * 16 + mrow;
  const long rowLd = rowA < N ? rowA : (long)(N - 1);
  const float* gA = in + rowLd * KIN;

  v8f acc[8] = {};                            // t<4: left tiles, t>=4: right
  for (int k0 = 0; k0 < KIN; k0 += 4) {
    v2f af = *(const v2f*)(gA + k0 + kOff);   // A[row][k0+kOff .. +1]
#pragma unroll
    for (int t = 0; t < 8; ++t) {
      const int proj = t >> 2;
      const float* wb = tW + proj * (KIN * NMEM) + (k0 << 6)
                      + ((((t & 3) << 4) + mrow) << 2) + kOff;
      v2f bf = *(const v2f*)wb;               // {W[k][n], W[k+1][n]}
      acc[t] = __builtin_amdgcn_wmma_f32_16x16x4_f32(
          false, af, false, bf, (short)0, acc[t], false, false);
    }
  }

  const int col  = lane & 15;
  const int rAdd = (lane >> 4) << 3;          // upper lanes hold rows M+8
#pragma unroll
  for (int nt = 0; nt < 4; ++nt) {
    const float bLv = bl[nt * 16 + col];
    const float bRv = br[nt * 16 + col];
#pragma unroll
    for (int e = 0; e < 8; ++e) {
      const long row = frame0 + wave * 16 + e + rAdd;
      if (row < N) {
        const float sl = sigm(acc[nt][e] + bLv);
        const float sr = sigm(acc[4 + nt][e] + bRv);
        Aarr[row * 64 + nt * 16 + col] = sl * sr;    // a = left*right
        Barr[row * 64 + nt * 16 + col] = 1.0f - sl;  // b = 1-left
      }
    }
  }
}

// -------------------------------------------------------------------------
// Kernel 2: per-chunk affine composition.  256 threads = 64 ch x 4 subs.
// -------------------------------------------------------------------------
extern "C" __global__ __launch_bounds__(256) void k_chunk(
    const float* __restrict__ Aarr, const float* __restrict__ Barr,
    float* __restrict__ Ps, float* __restrict__ Qs,
    float* __restrict__ Pc, float* __restrict__ Qc, int Nab)
{
  __shared__ float sP[4][NMEM];
  __shared__ float sQ[4][NMEM];
  const int tid = threadIdx.x;
  const int ch  = tid & 63;
  const int sub = tid >> 6;
  const long base = (long)blockIdx.x * CHUNK + (long)sub * SUB;

  float P = 1.0f, Q = 0.0f;
  for (int j = 0; j < SUB; ++j) {
    const long i = base + j;
    if (i < Nab) {
      if (i + 32 < Nab) {
        __builtin_prefetch(&Aarr[(i + 32) * 64 + ch], 0, 1);
        __builtin_prefetch(&Barr[(i + 32) * 64 + ch], 0, 1);
      }
      const float av = Aarr[i * 64 + ch];
      const float bv = Barr[i * 64 + ch];
      Q = av * Q + bv;          // compose in increasing i
      P = av * P;
    }
  }
  const size_t si = ((size_t)blockIdx.x * 4 + sub) * 64 + ch;
  Ps[si] = P;
  Qs[si] = Q;
  sP[sub][ch] = P;
  sQ[sub][ch] = Q;
  __syncthreads();
  if (sub == 0) {
    float cP = 1.0f, cQ = 0.0f;
#pragma unroll
    for (int s = 0; s < 4; ++s) {
      const float p = sP[s][ch], q = sQ[s][ch];
      cQ = p * cQ + q;
      cP = p * cP;
    }
    Pc[(size_t)blockIdx.x * 64 + ch] = cP;
    Qc[(size_t)blockIdx.x * 64 + ch] = cQ;
  }
}

// -------------------------------------------------------------------------
// Kernel 3: serial scan over chunk summaries. 64 threads, one per channel.
// -------------------------------------------------------------------------
extern "C" __global__ void k_scan(const float* __restrict__ Pc,
                                  const float* __restrict__ Qc,
                                  float* __restrict__ U0, int nchunks)
{
  const int ch = threadIdx.x;
  float u = 0.0f;
#pragma unroll 8
  for (int c = 0; c < nchunks; ++c) {
    U0[(size_t)c * 64 + ch] = u;
    u = Pc[(size_t)c * 64 + ch] * u + Qc[(size_t)c * 64 + ch];
  }
}

// -------------------------------------------------------------------------
// Kernel 4: apply scan -> z, written straight into d_out.
// -------------------------------------------------------------------------
extern "C" __global__ __launch_bounds__(256) void k_apply(
    const float* __restrict__ Aarr, const float* __restrict__ Barr,
    const float* __restrict__ Ps, const float* __restrict__ Qs,
    const float* __restrict__ U0, float* __restrict__ Z, int Nab, int N)
{
  const int tid = threadIdx.x;
  const int ch  = tid & 63;
  const int sub = tid >> 6;
  const long c  = blockIdx.x;

  float u = U0[c * 64 + ch];
  if (sub == 0) {
    const long t0 = c * CHUNK;
    if (t0 < N) Z[t0 * 64 + ch] = u;    // chunk-start z (covers z[0]=0)
  }
  for (int s = 0; s < sub; ++s) {       // advance to sub-chunk start
    const size_t si = (size_t)(c * 4 + s) * 64 + ch;
    u = Ps[si] * u + Qs[si];
  }
  const long base = c * CHUNK + (long)sub * SUB;
  for (int j = 0; j < SUB; ++j) {
    const long i = base + j;
    if (i >= Nab) break;
    u = Aarr[i * 64 + ch] * u + Barr[i * 64 + ch];
    Z[(i + 1) * 64 + ch] = u;
  }
}

// -------------------------------------------------------------------------
// Kernel 5: logits = z @ W_logit + b_logit, in place over d_out.
// z tile (128x64 f32) DMA'd into LDS by the Tensor Data Mover with
// pad_enable producing the stride-68 bank-spread layout; tensor_dim1 =
// remaining rows gives hardware-zeroed tail. W_logit quad-packed as in k1.
// -------------------------------------------------------------------------
extern "C" __global__ __launch_bounds__(256) void k_logits(
    float* __restrict__ Z, const float* __restrict__ Wlg,
    const float* __restrict__ blg, int N)
{
  __shared__ float tZ[128 * 68];        // TDM dest: 64 DW rows + 4 DW pad
  __shared__ float tW[64 * 64];         // quad-packed [k/4][n][k%4]
  const int tid = threadIdx.x;
  const long row0 = (long)blockIdx.x * 128;

  if (tid < 32) {                       // wave 0 issues the TDM transfer
    const unsigned long long gaddr =
        (unsigned long long)(uintptr_t)(Z + row0 * 64);
    const unsigned int ldsOff = (unsigned int)(uintptr_t)&tZ[0];
    const unsigned int rem = (unsigned int)(N - row0);   // rows left in tensor

    uint32x4 g0;
    g0[0] = 1u;                                          // count=1, user desc
    g0[1] = ldsOff;                                      // lds_addr
    g0[2] = (unsigned int)gaddr;                         // global_addr[95:64]
    g0[3] = ((unsigned int)(gaddr >> 32) & 0x01FFFFFFu)  // global_addr[120:96]
          | (2u << 30);                                  // type = 2 (image)

    int32x8 g1;
    g1[0] = (int)((2u << 16)        // data_size = 4 bytes
                | (1u << 20)        // pad_enable
                | (5u << 22)        // pad_interval: 2^5*8B = 256B = 64 DW
                | (3u << 25));      // pad_amount: 4 DWORDs -> stride 68
    g1[1] = (int)(64u << 16);       // tensor_dim0[15:0] = 64 (cols)
    g1[2] = (int)((rem & 0xFFFFu) << 16);          // tensor_dim1 lo
    g1[3] = (int)(((rem >> 16) & 0xFFFFu)          // tensor_dim1 hi
                | (64u << 16));                    // tile_dim0 = 64
    g1[4] = (int)128u;              // tile_dim1 = 128 rows
    g1[5] = (int)64u;               // tensor_dim0_stride = 64
    g1[6] = 0;
    g1[7] = 0;

    int32x4 gz4 = {0, 0, 0, 0};     // groups 2/3 unused (2-D tensor)
    int32x8 gz8 = {0, 0, 0, 0, 0, 0, 0, 0};
    __builtin_amdgcn_tensor_load_to_lds(g0, g1, gz4, gz4, gz8, 0);
    __builtin_amdgcn_s_wait_tensorcnt(0);
  }

  for (int idx = tid; idx < 64 * 64; idx += 256) {
    const int k = idx >> 6, n = idx & 63;
    tW[((k >> 2) << 8) + (n << 2) + (k & 3)] = Wlg[idx];
  }
  __syncthreads();

  const int lane = tid & 31;
  const int wave = tid >> 5;
  const int mrow = lane & 15;
  const int kOff = (lane >> 4) << 1;
  const float* zb = tZ + (wave * 16 + mrow) * 68 + kOff;

  v8f acc[4] = {};
  for (int k0 = 0; k0 < 64; k0 += 4) {
    v2f af = *(const v2f*)(zb + k0);
#pragma unroll
    for (int t = 0; t < 4; ++t) {
      const float* wb = tW + (k0 << 6) + ((((t & 3) << 4) + mrow) << 2) + kOff;
      v2f bf = *(const v2f*)wb;     // {W[k][n], W[k+1][n]}
      acc[t] = __builtin_amdgcn_wmma_f32_16x16x4_f32(
          false, af, false, bf, (short)0, acc[t], false, false);
    }
  }

  const int col  = lane & 15;
  const int rAdd = (lane >> 4) << 3;
#pragma unroll
  for (int t = 0; t < 4; ++t) {
    const float bv = blg[(t << 4) + col];
#pragma unroll
    for (int e = 0; e < 8; ++e) {
      const long row = row0 + wave * 16 + e + rAdd;
      if (row < N) Z[row * 64 + (t << 4) + col] = acc[t][e] + bv;
    }
  }
}

// -------------------------------------------------------------------------
// Host launcher
// -------------------------------------------------------------------------
extern "C" void kernel_launch(void* const* d_in, const int* in_sizes, int n_in,
                              void* d_out, int out_size, void* d_ws, size_t ws_size,
                              hipStream_t stream) {
  const float* in  = (const float*)d_in[0];
  // d_in[1] = targets (unused by the reference's returned value)
  const float* Wl  = (const float*)d_in[2];
  const float* bl  = (const float*)d_in[3];
  const float* Wr  = (const float*)d_in[4];
  const float* br  = (const float*)d_in[5];
  const float* Wlg = (const float*)d_in[6];
  const float* blg = (const float*)d_in[7];

  const int N   = in_sizes[0] / KIN;      // 500000
  const int Nab = N - 1;
  const int nchunks = (Nab + CHUNK - 1) / CHUNK;

  float* ws = (float*)d_ws;
  const size_t NM = (size_t)N * 64;
  float* A  = ws;
  float* B  = ws + NM;
  float* Ps = ws + 2 * NM;
  float* Qs = Ps + (size_t)nchunks * 4 * 64;
  float* Pc = Qs + (size_t)nchunks * 4 * 64;
  float* Qc = Pc + (size_t)nchunks * 64;
  float* U0 = Qc + (size_t)nchunks * 64;
  float* Z  = (float*)d_out;              // z lives in d_out, transformed in place

  const int gTiles = (N + 127) / 128;
  k_proj  <<<gTiles,  256, 0, stream>>>(in, Wl, bl, Wr, br, A, B, N);
  k_chunk <<<nchunks, 256, 0, stream>>>(A, B, Ps, Qs, Pc, Qc, Nab);
  k_scan  <<<1,        64, 0, stream>>>(Pc, Qc, U0, nchunks);
  k_apply <<<nchunks, 256, 0, stream>>>(A, B, Ps, Qs, U0, Z, Nab, N);
  k_logits<<<gTiles,  256, 0, stream>>>(Z, Wlg, blg, N);
}